// Model_60533269069830
// MI455X (gfx1250) — compile-verified
//
#include <hip/hip_runtime.h>
#include <stdint.h>

// Problem constants (from reference)
#define POOL 8192
#define W    8192
#define NB   16
#define MAXP 4096
#define MAXE 2048

#define ROWS_PER_BLOCK 2
#define NTHREADS 256
#define CHUNKS (W / (NTHREADS * 4))   // 8 x b128 per thread per row

// ---- CDNA5 async global<->LDS DMA helpers (ASYNCcnt path) -------------------
// load:  global_load_async_to_lds_b128  vdst(lds addr), vaddr(64b global), off
// store: global_store_async_from_lds_b128 vaddr(64b global), vsrc(lds addr), off
__device__ __forceinline__ void async_load_b128_nt(unsigned lds_addr, const void* gptr) {
    asm volatile("global_load_async_to_lds_b128 %0, %1, off th:TH_LOAD_NT"
                 :
                 : "v"(lds_addr), "v"((unsigned long long)(uintptr_t)gptr)
                 : "memory");
}
__device__ __forceinline__ void async_store_b128_nt(void* gptr, unsigned lds_addr) {
    asm volatile("global_store_async_from_lds_b128 %0, %1, off th:TH_STORE_NT"
                 :
                 : "v"((unsigned long long)(uintptr_t)gptr), "v"(lds_addr)
                 : "memory");
}
__device__ __forceinline__ void wait_async0() {
    asm volatile("s_wait_asynccnt 0x0" ::: "memory");
}

// Two table rows per block. Unmodified rows: staged through LDS with the async
// DMA engine (no VGPR traffic, 8 KB outstanding per wave before the single
// ASYNCcnt drain). Modified rows (16 of 8192): reference's
// where(pos<pre, prefix, where(pos<seq, ext, existing)) — issued while the
// async loads of the sibling row are in flight.
__global__ __launch_bounds__(NTHREADS) void row_writeback_kernel(
    const float* __restrict__ req_to_token,     // [POOL, W]
    const float* __restrict__ prefix_tensors,   // [NB, MAXP]
    const float* __restrict__ out_cache_loc,    // [NB*MAXE]
    const int*   __restrict__ req_pool_indices, // [NB]
    const int*   __restrict__ prefix_lens,      // [NB]
    const int*   __restrict__ seq_lens,         // [NB]
    const int*   __restrict__ extend_lens,      // [NB]
    float*       __restrict__ out)              // [POOL, W]
{
    __shared__ float smem[ROWS_PER_BLOCK][W];   // 64 KB: two full rows

    const int tid  = threadIdx.x;
    const int row0 = blockIdx.x * ROWS_PER_BLOCK;

    // Uniform per-block: which (if any) batch index owns each of our rows?
    int bsel[ROWS_PER_BLOCK];
#pragma unroll
    for (int r = 0; r < ROWS_PER_BLOCK; ++r) {
        int b = -1;
#pragma unroll
        for (int i = 0; i < NB; ++i) {
            if (req_pool_indices[i] == row0 + r) b = i;
        }
        bsel[r] = b;
    }

    // Low 32 bits of the generic pointer to a __shared__ object == LDS
    // logical byte address (flat shared aperture keeps it in addr[31:0]).
    unsigned ldsb[ROWS_PER_BLOCK];
#pragma unroll
    for (int r = 0; r < ROWS_PER_BLOCK; ++r)
        ldsb[r] = (unsigned)(uintptr_t)(void*)&smem[r][0];

    // ---- Phase 1: launch async loads for all unmodified rows (EXEC full) ----
#pragma unroll
    for (int r = 0; r < ROWS_PER_BLOCK; ++r) {
        if (bsel[r] < 0) {
            const float* __restrict__ src = req_to_token + (size_t)(row0 + r) * W;
#pragma unroll
            for (int k = 0; k < CHUNKS; ++k) {
                const int e = (k * NTHREADS + tid) * 4;   // float index in row
                async_load_b128_nt(ldsb[r] + e * 4, src + e);
            }
        }
    }

    // ---- Phase 2: modified-row VALU path rides under the DMA latency ----
#pragma unroll
    for (int r = 0; r < ROWS_PER_BLOCK; ++r) {
        const int b = bsel[r];
        if (b >= 0) {
            const float* __restrict__ src = req_to_token + (size_t)(row0 + r) * W;
            float*       __restrict__ dst = out          + (size_t)(row0 + r) * W;
            const int pre = prefix_lens[b];
            const int seq = seq_lens[b];
            int start = 0;                                // exclusive cumsum(extend_lens)
#pragma unroll
            for (int i = 0; i < NB; ++i) {
                if (i < b) start += extend_lens[i];
            }
            for (int j = tid; j < W; j += NTHREADS) {
                float v;
                if (j < pre) {
                    const int pj = (j < MAXP - 1) ? j : (MAXP - 1);
                    v = prefix_tensors[b * MAXP + pj];
                } else if (j < seq) {
                    int idx = start + (j - pre);
                    if (idx < 0) idx = 0;
                    if (idx > NB * MAXE - 1) idx = NB * MAXE - 1;
                    v = out_cache_loc[idx];
                } else {
                    v = src[j];                            // original table value
                }
                dst[j] = v;
            }
        }
    }

    // ---- Phase 3: single per-wave drain, then push both rows back out ----
    wait_async0();
#pragma unroll
    for (int r = 0; r < ROWS_PER_BLOCK; ++r) {
        if (bsel[r] < 0) {
            float* __restrict__ dst = out + (size_t)(row0 + r) * W;
#pragma unroll
            for (int k = 0; k < CHUNKS; ++k) {
                const int e = (k * NTHREADS + tid) * 4;
                async_store_b128_nt(dst + e, ldsb[r] + e * 4);
            }
        }
    }
    wait_async0();   // drain before kernel end
}

extern "C" void kernel_launch(void* const* d_in, const int* in_sizes, int n_in,
                              void* d_out, int out_size, void* d_ws, size_t ws_size,
                              hipStream_t stream) {
    const float* req_to_token     = (const float*)d_in[0];
    const float* prefix_tensors   = (const float*)d_in[1];
    const float* out_cache_loc    = (const float*)d_in[2];
    const int*   req_pool_indices = (const int*)d_in[3];
    const int*   prefix_lens      = (const int*)d_in[4];
    const int*   seq_lens         = (const int*)d_in[5];
    const int*   extend_lens      = (const int*)d_in[6];
    float*       out              = (float*)d_out;

    (void)in_sizes; (void)n_in; (void)out_size; (void)d_ws; (void)ws_size;

    row_writeback_kernel<<<dim3(POOL / ROWS_PER_BLOCK), dim3(NTHREADS), 0, stream>>>(
        req_to_token, prefix_tensors, out_cache_loc,
        req_pool_indices, prefix_lens, seq_lens, extend_lens, out);
}